// AdaptiveEmbeddingGraphBuilder_72602127171697
// MI455X (gfx1250) — compile-verified
//
#include <hip/hip_runtime.h>

// MI455X / gfx1250, wave32. Two kernels:
//  K1: A = relu(E E^T) via V_WMMA_F32_16X16X4_F32, streamed to d_out.
//  K2: per-row top-10 threshold + stable softmax, in place on d_out.
// No workspace needed: d_out (8192x8192 f32) is the scratch.

#define N_NODES 8192
#define EMBED   64
#define TOPK    10

typedef float v2f __attribute__((ext_vector_type(2)));
typedef float v8f __attribute__((ext_vector_type(8)));

// ---------------- Kernel 1: GEMM + relu --------------------------------
// Block = 256 threads (8 waves). Block owns a 16-row strip; each wave walks
// column tiles jt = wave, wave+8, ... (64 tiles of 16 cols each).
// f32 WMMA fragment mapping (ISA 7.12.2, 32-bit A 16x4 / B 4x16):
//   A[m][k]: lane = (k/2)*16 + m, vgpr = k%2
//   B[k][n]: lane = (k/2)*16 + n, vgpr = k%2
// -> per step s (k base 4s), lane loads E[row + lane%16][4s + 2*(lane/16) + {0,1}]
__global__ __launch_bounds__(256)
void gemm_relu_kernel(const float* __restrict__ E, float* __restrict__ out) {
    const int lane  = threadIdx.x & 31;
    const int wave  = threadIdx.x >> 5;
    const int i0    = blockIdx.x << 4;       // row strip base
    const int mrow  = lane & 15;             // m (for A) / n (for B)
    const int khalf = lane >> 4;

    // A fragment for this 16-row strip: reused for all 512 column tiles.
    v2f afrag[16];
    const float* arow = E + (i0 + mrow) * EMBED + 2 * khalf;
#pragma unroll
    for (int s = 0; s < 16; ++s)
        afrag[s] = *(const v2f*)(arow + 4 * s);

    for (int jt = wave; jt < N_NODES / 16; jt += 8) {
        const int j0 = jt << 4;
        const float* brow = E + (j0 + mrow) * EMBED + 2 * khalf;
        v2f bfrag[16];
#pragma unroll
        for (int s = 0; s < 16; ++s)
            bfrag[s] = *(const v2f*)(brow + 4 * s);

        v8f acc = {0.f, 0.f, 0.f, 0.f, 0.f, 0.f, 0.f, 0.f};
#pragma unroll
        for (int s = 0; s < 16; ++s) {
            // 8 args: (neg_a, A, neg_b, B, c_mod, C, reuse_a, reuse_b)
            acc = __builtin_amdgcn_wmma_f32_16x16x4_f32(
                false, afrag[s], false, bfrag[s], (short)0, acc, false, false);
        }

        // C/D layout: vgpr q, lane L -> col = L%16, row = q + 8*(L/16)
        const int col = j0 + mrow;
#pragma unroll
        for (int q = 0; q < 8; ++q) {
            float v = acc[q];
            v = v > 0.f ? v : 0.f;  // relu
            out[(size_t)(i0 + q + 8 * khalf) * N_NODES + col] = v;
        }
    }
}

// ---------------- Kernel 2: per-row top-10 + softmax -------------------
__device__ __forceinline__ float wave_max_f(float x) {
    x = fmaxf(x, __int_as_float(__builtin_amdgcn_ds_swizzle(__float_as_int(x), (16 << 10) | 31)));
    x = fmaxf(x, __int_as_float(__builtin_amdgcn_ds_swizzle(__float_as_int(x), ( 8 << 10) | 31)));
    x = fmaxf(x, __int_as_float(__builtin_amdgcn_ds_swizzle(__float_as_int(x), ( 4 << 10) | 31)));
    x = fmaxf(x, __int_as_float(__builtin_amdgcn_ds_swizzle(__float_as_int(x), ( 2 << 10) | 31)));
    x = fmaxf(x, __int_as_float(__builtin_amdgcn_ds_swizzle(__float_as_int(x), ( 1 << 10) | 31)));
    return x;
}
__device__ __forceinline__ int wave_min_i(int x) {
    x = min(x, __builtin_amdgcn_ds_swizzle(x, (16 << 10) | 31));
    x = min(x, __builtin_amdgcn_ds_swizzle(x, ( 8 << 10) | 31));
    x = min(x, __builtin_amdgcn_ds_swizzle(x, ( 4 << 10) | 31));
    x = min(x, __builtin_amdgcn_ds_swizzle(x, ( 2 << 10) | 31));
    x = min(x, __builtin_amdgcn_ds_swizzle(x, ( 1 << 10) | 31));
    return x;
}

// Block = 256 threads = 8 waves; one wave per row. Grid = 8192/8 = 1024.
__global__ __launch_bounds__(256)
void topk_softmax_kernel(float* __restrict__ out) {
    const int lane = threadIdx.x & 31;
    const int wave = threadIdx.x >> 5;
    const int row  = (blockIdx.x << 3) + wave;
    float* rowp = out + (size_t)row * N_NODES;
    const float4* row4 = (const float4*)rowp;

    // Phase 1: lane-local sorted (descending) top-10, branchless insertion:
    //   t'[0] = max(t[0], v);  t'[q] = median(v, t[q], t[q-1]) = min(max(v,t[q]), t[q-1])
    float t[10];
#pragma unroll
    for (int q = 0; q < 10; ++q) t[q] = -1.0f;  // relu output >= 0

    for (int it = 0; it < N_NODES / (4 * 32); ++it) {
        float4 v4 = row4[lane + it * 32];
        float vs[4] = {v4.x, v4.y, v4.z, v4.w};
#pragma unroll
        for (int e = 0; e < 4; ++e) {
            float v = vs[e];
#pragma unroll
            for (int q = 9; q >= 1; --q)
                t[q] = fminf(fmaxf(v, t[q]), t[q - 1]);
            t[0] = fmaxf(t[0], v);
        }
    }

    // Phase 2: cross-lane merge by popping 10 global maxima.
    float M = 0.f, sumExp = 0.f, thr = 0.f;
#pragma unroll
    for (int k = 0; k < 10; ++k) {
        float g = wave_max_f(t[0]);   // lists sorted desc -> head is lane max
        if (k == 0) M = g;            // row max (softmax stabilizer)
        sumExp += __expf(g - M);
        thr = g;                      // after loop: 10th largest
        int owner = wave_min_i((t[0] == g) ? lane : 64);
        bool isO = (lane == owner);   // lowest lane holding g pops its head
#pragma unroll
        for (int q = 0; q < 9; ++q) t[q] = isO ? t[q + 1] : t[q];
        t[9] = isO ? -1.0f : t[9];
    }

    // Masked entries contribute exp(0 - M); N-TOPK of them.
    const float denom = sumExp + (float)(N_NODES - TOPK) * __expf(-M);
    const float inv   = 1.0f / denom;
    const float eneg  = __expf(-M) * inv;   // value for non-top-k entries

    // Phase 3: rewrite the row in place (rows are cache-hot: 32KB).
    for (int it = 0; it < N_NODES / (4 * 32); ++it) {
        float4 v4 = row4[lane + it * 32];
        float4 o;
        o.x = (v4.x >= thr) ? __expf(v4.x - M) * inv : eneg;
        o.y = (v4.y >= thr) ? __expf(v4.y - M) * inv : eneg;
        o.z = (v4.z >= thr) ? __expf(v4.z - M) * inv : eneg;
        o.w = (v4.w >= thr) ? __expf(v4.w - M) * inv : eneg;
        ((float4*)rowp)[lane + it * 32] = o;
    }
}

extern "C" void kernel_launch(void* const* d_in, const int* in_sizes, int n_in,
                              void* d_out, int out_size, void* d_ws, size_t ws_size,
                              hipStream_t stream) {
    (void)in_sizes; (void)n_in; (void)out_size; (void)d_ws; (void)ws_size;
    const float* E = (const float*)d_in[0];
    float* out = (float*)d_out;

    gemm_relu_kernel<<<N_NODES / 16, 256, 0, stream>>>(E, out);
    topk_softmax_kernel<<<N_NODES / 8, 256, 0, stream>>>(out);
}